// RelationalEdgeDistributionDecoder_1855425872454
// MI455X (gfx1250) — compile-verified
//
#include <hip/hip_runtime.h>
#include <hip/hip_bf16.h>

typedef __attribute__((ext_vector_type(2))) float v2f;
typedef __attribute__((ext_vector_type(8))) float v8f;

#define D_DIM 128

// ---------------------------------------------------------------------------
// Kernel 1: gene_proj = leaky_relu(gene_z @ W_gene + b_gene)
// One block (256 thr = 8 wave32) per 16 gene rows. Wave w computes the
// 16x16 output tile at columns [16w, 16w+16) using V_WMMA_F32_16X16X4_F32,
// K=128 -> 32 WMMA ops per wave.
// ---------------------------------------------------------------------------
__global__ void __launch_bounds__(256)
gene_proj_wmma_kernel(const float* __restrict__ gene_z,
                      const float* __restrict__ W,
                      const float* __restrict__ bvec,
                      float* __restrict__ proj,
                      int n_gene)
{
    // LDS A-tile: 16 rows x 128 cols, row stride 131 floats (conflict-free
    // for the half-wave column read pattern: 131 % 64 == 3).
    __shared__ float As[16][131];

    const int tid  = threadIdx.x;
    const int row0 = blockIdx.x * 16;

    // Stage A tile (16x128 = 2048 floats, 8 per thread), coalesced global reads.
    #pragma unroll
    for (int i = tid; i < 16 * D_DIM; i += 256) {
        int r = i >> 7;          // i / 128
        int c = i & 127;         // i % 128
        int rr = row0 + r;
        if (rr >= n_gene) rr = n_gene - 1;   // clamp for a partial last tile
        As[r][c] = gene_z[(long)rr * D_DIM + c];
    }
    __syncthreads();

    const int wave = tid >> 5;
    const int lane = tid & 31;
    const int half = lane >> 4;        // 0: K pair {0,1}; 1: K pair {2,3}
    const int l16  = lane & 15;
    const int col0 = wave * 16;
    const int n    = col0 + l16;       // output column for this lane

    v8f acc = {};

    #pragma unroll 8
    for (int kk = 0; kk < D_DIM / 4; ++kk) {
        const int kb = kk * 4 + half * 2;
        // A fragment (16x4): lane L -> M = L&15, VGPR0 = K=kb, VGPR1 = K=kb+1
        v2f a;
        a[0] = As[l16][kb];
        a[1] = As[l16][kb + 1];
        // B fragment (4x16): lane L -> N = col0 + (L&15), rows striped:
        // VGPR0 = K=kb, VGPR1 = K=kb+1 (lanes>=16 carry K+2 via `half`)
        v2f b;
        b[0] = W[(long)kb * D_DIM + n];
        b[1] = W[(long)(kb + 1) * D_DIM + n];

        acc = __builtin_amdgcn_wmma_f32_16x16x4_f32(
            /*neg_a=*/false, a, /*neg_b=*/false, b,
            /*c_mod=*/(short)0, acc, /*reuse_a=*/false, /*reuse_b=*/false);
    }

    const float bias = bvec[n];

    if (row0 + 16 <= n_gene) {
        // Full tile: branch-free epilogue (uniform path for all full blocks).
        #pragma unroll
        for (int j = 0; j < 8; ++j) {
            const int m = j + half * 8;        // C/D layout: VGPR j -> row
            float v = acc[j] + bias;
            v = (v > 0.0f) ? v : 0.01f * v;    // leaky_relu, slope 0.01
            proj[(long)(row0 + m) * D_DIM + n] = v;
        }
    } else {
        // Partial last tile only.
        #pragma unroll
        for (int j = 0; j < 8; ++j) {
            const int m = j + half * 8;
            float v = acc[j] + bias;
            v = (v > 0.0f) ? v : 0.01f * v;
            const int rr = row0 + m;
            if (rr < n_gene)
                proj[(long)rr * D_DIM + n] = v;
        }
    }
}

// ---------------------------------------------------------------------------
// Kernel 2: per-edge gather + 128-dot + distribution params.
// One wave32 per edge: each lane loads a float4 of u and v (512B coalesced
// row reads from L2-resident tables), xor-shuffle wave32 reduction, lane 0
// epilogue. Row indices are forced into SGPRs (readfirstlane) so the row
// gathers use the SADDR + lane-offset global_load_b128 form.
// ---------------------------------------------------------------------------
__global__ void __launch_bounds__(256)
edge_dot_kernel(const float* __restrict__ cell_z,
                const float* __restrict__ gene_proj,
                const float* __restrict__ cell_scale,
                const float* __restrict__ cell_bias,
                const float* __restrict__ cell_std,
                const float* __restrict__ gene_scale,
                const float* __restrict__ gene_bias,
                const float* __restrict__ gene_std,
                const int* __restrict__ src_idx,
                const int* __restrict__ dst_idx,
                const int* __restrict__ cell_n_id,
                const int* __restrict__ gene_n_id,
                float* __restrict__ out,
                int E)
{
    const int lane = threadIdx.x & 31;
    const long e = (long)blockIdx.x * (blockDim.x >> 5) + (threadIdx.x >> 5);
    if (e >= E) return;

    // Wave-uniform edge endpoints -> SGPRs.
    const int s = __builtin_amdgcn_readfirstlane(src_idx[e]);
    const int d = __builtin_amdgcn_readfirstlane(dst_idx[e]);

    const float4 u4 = *(const float4*)(cell_z    + (long)s * D_DIM + lane * 4);
    const float4 v4 = *(const float4*)(gene_proj + (long)d * D_DIM + lane * 4);

    float p = u4.x * v4.x + u4.y * v4.y + u4.z * v4.z + u4.w * v4.w;

    // wave32 reduction
    #pragma unroll
    for (int off = 16; off > 0; off >>= 1)
        p += __shfl_xor(p, off, 32);

    if (lane == 0) {
        const int sid = cell_n_id[s];
        const int did = gene_n_id[d];
        const float loc = p * cell_scale[sid] * gene_scale[did]
                          + cell_bias[sid] + gene_bias[did];
        const float sd  = cell_std[sid] * gene_std[did];
        out[e]           = loc;   // row 0 of [2, E]
        out[(long)E + e] = sd;    // row 1 of [2, E]
    }
}

// ---------------------------------------------------------------------------
// Launch
// ---------------------------------------------------------------------------
extern "C" void kernel_launch(void* const* d_in, const int* in_sizes, int n_in,
                              void* d_out, int out_size, void* d_ws, size_t ws_size,
                              hipStream_t stream) {
    const float* cell_z     = (const float*)d_in[0];
    const float* gene_z     = (const float*)d_in[1];
    const float* W_gene     = (const float*)d_in[2];
    const float* b_gene     = (const float*)d_in[3];
    const float* cell_scale = (const float*)d_in[4];
    const float* cell_bias  = (const float*)d_in[5];
    const float* cell_std   = (const float*)d_in[6];
    const float* gene_scale = (const float*)d_in[7];
    const float* gene_bias  = (const float*)d_in[8];
    const float* gene_std   = (const float*)d_in[9];
    const int*   src_idx    = (const int*)d_in[10];
    const int*   dst_idx    = (const int*)d_in[11];
    const int*   cell_n_id  = (const int*)d_in[12];
    const int*   gene_n_id  = (const int*)d_in[13];

    const int n_gene = in_sizes[1] / D_DIM;   // 20000
    const int E      = in_sizes[10];          // 1,000,000

    float* gene_proj = (float*)d_ws;          // n_gene * 128 floats (10.24 MB)

    const int gemm_blocks = (n_gene + 15) / 16;
    gene_proj_wmma_kernel<<<gemm_blocks, 256, 0, stream>>>(
        gene_z, W_gene, b_gene, gene_proj, n_gene);

    const int edges_per_block = 256 / 32;     // 8 waves -> 8 edges per block
    const int edge_blocks = (E + edges_per_block - 1) / edges_per_block;
    edge_dot_kernel<<<edge_blocks, 256, 0, stream>>>(
        cell_z, gene_proj,
        cell_scale, cell_bias, cell_std,
        gene_scale, gene_bias, gene_std,
        src_idx, dst_idx, cell_n_id, gene_n_id,
        (float*)d_out, E);
}